// SQNet_40965398069607
// MI455X (gfx1250) — compile-verified
//
#include <hip/hip_runtime.h>
#include <math.h>

typedef float v2f __attribute__((ext_vector_type(2)));
typedef float v8f __attribute__((ext_vector_type(8)));

#ifndef M_PI
#define M_PI 3.14159265358979323846
#endif

// Fast base-2 transcendentals -> v_exp_f32 / v_log_f32
__device__ __forceinline__ float fast_exp2(float x) {
#if __has_builtin(__builtin_amdgcn_exp2f)
    return __builtin_amdgcn_exp2f(x);
#else
    return exp2f(x);
#endif
}
__device__ __forceinline__ float fast_log2(float x) {
#if __has_builtin(__builtin_amdgcn_logf)
    return __builtin_amdgcn_logf(x);   // v_log_f32 is log2
#else
    return __log2f(x);
#endif
}
// pow for strictly-positive base
__device__ __forceinline__ float fast_pow(float x, float y) {
    return fast_exp2(y * fast_log2(x));
}

// ---------------------------------------------------------------------------
// Weight-normalized linear layer (batch=1 GEMV) via V_WMMA_F32_16X16X4_F32.
// One wave per block, one block per 16-output tile.
//   y = V(16 x fin) . x(fin)   accumulated in the M=0 row of the 16x16 C tile
//   A(16x4): row0 = x chunk (lane0 holds K0,K1; lane16 holds K2,K3), rest 0.
//            Branch-free: every lane loads its float2 of x, selects on col==0.
//   B(4x16): lane n<16 holds B[0..1][n] = V[row0+n][kb+0..1]
//            lane n>=16 holds B[2..3][n-16] = V[row0+n-16][kb+2..3]
//   Two accumulators (even/odd chunks) break the WMMA C-chain; summed at end.
//   ssq (||v_row||^2) accumulated from fp32 weights in the same loop.
// ---------------------------------------------------------------------------
__global__ __launch_bounds__(32)
void wn_layer_wmma(const float* __restrict__ x, const float* __restrict__ V,
                   const float* __restrict__ g, const float* __restrict__ b,
                   float* __restrict__ out, int fin, int fout,
                   int use_wn, int use_relu, int out_offset)
{
    const int lane = threadIdx.x;        // 0..31
    const int row0 = blockIdx.x * 16;    // output tile base
    const int col  = lane & 15;          // weight row within tile
    const int hi   = lane >> 4;          // which K-pair this lane supplies
    const bool isc0 = (col == 0);

    v8f c0 = {};
    v8f c1 = {};
    float ssq = 0.0f;

    const float* vrowh = V + (size_t)(row0 + col) * (size_t)fin + 2 * hi;
    const float* xbase = x + 2 * hi;

    // fin is a multiple of 8 (256 or 512)
    #pragma unroll 4
    for (int kb = 0; kb < fin; kb += 8) {
        v2f xv0 = *(const v2f*)(xbase + kb);
        v2f xv1 = *(const v2f*)(xbase + kb + 4);
        v2f b0  = *(const v2f*)(vrowh + kb);
        v2f b1  = *(const v2f*)(vrowh + kb + 4);

        v2f a0, a1;
        a0.x = isc0 ? xv0.x : 0.0f;
        a0.y = isc0 ? xv0.y : 0.0f;
        a1.x = isc0 ? xv1.x : 0.0f;
        a1.y = isc0 ? xv1.y : 0.0f;

        ssq += b0.x * b0.x + b0.y * b0.y;
        ssq += b1.x * b1.x + b1.y * b1.y;

        c0 = __builtin_amdgcn_wmma_f32_16x16x4_f32(
                 false, a0, false, b0, (short)0, c0, false, false);
        c1 = __builtin_amdgcn_wmma_f32_16x16x4_f32(
                 false, a1, false, b1, (short)0, c1, false, false);
    }

    // combine the two K-half partial sums of ||v_row||^2
    ssq += __shfl_xor(ssq, 16, 32);

    if (lane < 16) {
        int r = row0 + lane;
        if (r < fout) {
            float y = c0[0] + c1[0];              // D[M=0][N=lane]
            float o;
            if (use_wn) o = g[r] * y / sqrtf(ssq) + b[r];
            else        o = y + b[r];
            if (use_relu) o = fmaxf(o, 0.0f);
            out[out_offset + r] = o;
        }
    }
}

// ---------------------------------------------------------------------------
__global__ void copy_vec(const float* __restrict__ src, float* __restrict__ dst, int n)
{
    int i = blockIdx.x * blockDim.x + threadIdx.x;
    if (i < n) dst[i] = src[i];
}

// ---------------------------------------------------------------------------
// sigmoid + scale/shift -> sp ; then per-SQ derived constants (20 floats each):
//  [0..2]=t, [3..11]=R rowmajor, [12..14]=1/size, [15]=2/e2, [16]=e2/e1,
//  [17]=2/e1, [18]=e1/2, [19]=min(size)
// ---------------------------------------------------------------------------
#define SQF 20

__global__ __launch_bounds__(384)
void finalize_params(const float* __restrict__ raw, float* __restrict__ sp,
                     float* __restrict__ out_sp, float* __restrict__ sqd,
                     int outdim, int K)
{
    int i = threadIdx.x;
    if (i < outdim) {
        float r = raw[i];
        float s = 1.0f / (1.0f + __expf(-r));
        int p = i % 11;
        float scale, shift;
        if (p < 3)      { scale = 1.0f;                shift = -0.5f; }
        else if (p < 6) { scale = 2.0f * (float)M_PI;  shift = -(float)M_PI; }
        else if (p < 9) { scale = 0.4f;                shift = 0.1f; }
        else            { scale = 0.99f;               shift = 0.01f; }
        float v = s * scale + shift;
        sp[i] = v;
        out_sp[i] = v;
    }
    __syncthreads();
    if (i < K) {
        const float* q = sp + i * 11;
        float* d = sqd + i * SQF;
        d[0] = q[0]; d[1] = q[1]; d[2] = q[2];

        float ax = q[3], ay = q[4], az = q[5];
        float cx = cosf(ax), sx = sinf(ax);
        float cy = cosf(ay), sy = sinf(ay);
        float cz = cosf(az), sz = sinf(az);
        float Rx[9] = {1,0,0,  0,cx,-sx,  0,sx,cx};
        float Ry[9] = {cy,0,sy,  0,1,0,  -sy,0,cy};
        float Rz[9] = {cz,-sz,0,  sz,cz,0,  0,0,1};
        float M[9], R[9];
        for (int r2 = 0; r2 < 3; ++r2)
            for (int c2 = 0; c2 < 3; ++c2) {
                float acc = 0.0f;
                for (int k = 0; k < 3; ++k) acc += Ry[r2*3+k] * Rx[k*3+c2];
                M[r2*3+c2] = acc;
            }
        for (int r2 = 0; r2 < 3; ++r2)
            for (int c2 = 0; c2 < 3; ++c2) {
                float acc = 0.0f;
                for (int k = 0; k < 3; ++k) acc += Rz[r2*3+k] * M[k*3+c2];
                R[r2*3+c2] = acc;
            }
        for (int k = 0; k < 9; ++k) d[3+k] = R[k];

        float s0 = q[6], s1 = q[7], s2 = q[8];
        d[12] = 1.0f / s0; d[13] = 1.0f / s1; d[14] = 1.0f / s2;
        float e1 = fmaxf(q[9],  1e-4f);
        float e2 = fmaxf(q[10], 1e-4f);
        d[15] = 2.0f / e2;
        d[16] = e2 / e1;
        d[17] = 2.0f / e1;
        d[18] = e1 * 0.5f;
        d[19] = fminf(s0, fminf(s1, s2));
    }
}

// ---------------------------------------------------------------------------
// SDF: wave -> point, lane -> superquadric (K == 32 == wave32).
// pow(x,y) = exp2(y*log2(x)) with x>0 guaranteed (abs + 1e-6).
// Stores: each wave writes 128B contiguous (fully coalesced).
// ---------------------------------------------------------------------------
__global__ __launch_bounds__(256)
void sdf_kernel(const float* __restrict__ qp, const float* __restrict__ sqd,
                float* __restrict__ out, int N, int K)
{
    __shared__ float lds[32 * SQF];
    int tid = threadIdx.x;
    for (int i = tid; i < K * SQF; i += blockDim.x) lds[i] = sqd[i];
    __syncthreads();

    int lane = tid & 31;
    int n = blockIdx.x * (blockDim.x >> 5) + (tid >> 5);
    if (n >= N) return;

    float px = qp[3 * (size_t)n + 0];
    float py = qp[3 * (size_t)n + 1];
    float pz = qp[3 * (size_t)n + 2];

    if (lane < K) {
        const float* s = lds + lane * SQF;
        float d0 = px - s[0], d1 = py - s[1], d2 = pz - s[2];
        float r0 = s[3] * d0 + s[4]  * d1 + s[5]  * d2;
        float r1 = s[6] * d0 + s[7]  * d1 + s[8]  * d2;
        float r2 = s[9] * d0 + s[10] * d1 + s[11] * d2;
        float a0 = fabsf(r0 * s[12]) + 1e-6f;
        float a1 = fabsf(r1 * s[13]) + 1e-6f;
        float a2 = fabsf(r2 * s[14]) + 1e-6f;
        float pe2 = s[15], r21 = s[16], pe1 = s[17], eh = s[18];
        float t1 = fast_pow(a0, pe2) + fast_pow(a1, pe2);
        float t2 = fast_pow(t1, r21) + fast_pow(a2, pe1);
        float inside = fast_pow(t2, eh);
        out[(size_t)n * K + lane] = (inside - 1.0f) * s[19];
    }
}

// ---------------------------------------------------------------------------
extern "C" void kernel_launch(void* const* d_in, const int* in_sizes, int n_in,
                              void* d_out, int out_size, void* d_ws, size_t ws_size,
                              hipStream_t stream)
{
    // Input ordering: insertion order (z, qp, params...) vs pytree-sorted
    // (params..., qp, z). Detect via first array size.
    int iz, iqp, ip0;
    if (in_sizes[0] == 256) { iz = 0; iqp = 1; ip0 = 2; }
    else                    { ip0 = 0; iqp = n_in - 2; iz = n_in - 1; }

    const float* z  = (const float*)d_in[iz];
    const float* qp = (const float*)d_in[iqp];
    int N = in_sizes[iqp] / 3;

    // params: net1 (v,g,b)x4 | net2 (v,g,b)x4 | out (w,b)
    const float* P[26];
    for (int i = 0; i < 26; ++i) P[i] = (const float*)d_in[ip0 + i];
    int outdim = in_sizes[ip0 + 25];     // K*11 = 352
    int K = outdim / 11;                 // 32

    float* ws   = (float*)d_ws;
    float* bufA = ws;           // 512
    float* bufB = ws + 512;     // 512
    float* raw  = ws + 1024;    // outdim (<=384)
    float* sp   = ws + 1408;    // outdim
    float* sqd  = ws + 1792;    // K*SQF
    float* outf = (float*)d_out;

    // net1
    hipLaunchKernelGGL(wn_layer_wmma, dim3(512/16), dim3(32), 0, stream,
                       z,    P[0], P[1], P[2],  bufA, 256, 512, 1, 1, 0);
    hipLaunchKernelGGL(wn_layer_wmma, dim3(512/16), dim3(32), 0, stream,
                       bufA, P[3], P[4], P[5],  bufB, 512, 512, 1, 1, 0);
    hipLaunchKernelGGL(wn_layer_wmma, dim3(512/16), dim3(32), 0, stream,
                       bufB, P[6], P[7], P[8],  bufA, 512, 512, 1, 1, 0);
    hipLaunchKernelGGL(wn_layer_wmma, dim3(256/16), dim3(32), 0, stream,
                       bufA, P[9], P[10], P[11], bufB, 512, 256, 1, 1, 0);
    // concat: bufB[256..511] = z
    hipLaunchKernelGGL(copy_vec, dim3(1), dim3(256), 0, stream, z, bufB + 256, 256);
    // net2
    hipLaunchKernelGGL(wn_layer_wmma, dim3(512/16), dim3(32), 0, stream,
                       bufB, P[12], P[13], P[14], bufA, 512, 512, 1, 1, 0);
    hipLaunchKernelGGL(wn_layer_wmma, dim3(512/16), dim3(32), 0, stream,
                       bufA, P[15], P[16], P[17], bufB, 512, 512, 1, 1, 0);
    hipLaunchKernelGGL(wn_layer_wmma, dim3(512/16), dim3(32), 0, stream,
                       bufB, P[18], P[19], P[20], bufA, 512, 512, 1, 1, 0);
    hipLaunchKernelGGL(wn_layer_wmma, dim3(512/16), dim3(32), 0, stream,
                       bufA, P[21], P[22], P[23], bufB, 512, 512, 1, 1, 0);
    // out layer: plain linear (no weight norm, no relu)
    hipLaunchKernelGGL(wn_layer_wmma, dim3((outdim + 15)/16), dim3(32), 0, stream,
                       bufB, P[24], (const float*)nullptr, P[25], raw,
                       512, outdim, 0, 0, 0);

    // sigmoid + scale/shift -> sp ; derived per-SQ constants
    hipLaunchKernelGGL(finalize_params, dim3(1), dim3(384), 0, stream,
                       raw, sp, outf + (size_t)N * K, sqd, outdim, K);

    // SDF over all points
    int waves_per_block = 256 / 32;
    int blocks = (N + waves_per_block - 1) / waves_per_block;
    hipLaunchKernelGGL(sdf_kernel, dim3(blocks), dim3(256), 0, stream,
                       qp, sqd, outf, N, K);
}